// ImbalancedNoiseTopKLoss_14078902796490
// MI455X (gfx1250) — compile-verified
//
#include <hip/hip_runtime.h>
#include <hip/hip_bf16.h>
#include <math.h>

// ---------------------------------------------------------------------------
// ImbalancedNoiseTopKLoss for MI455X (gfx1250)
//
// skp1[b] = mean_n( 6th-largest_d( s[b,d] + 0.1*Z[b,d,n] ) )
// out     = mean_b( relu( 30*( m_list[y[b]] + skp1[b] - s[b,y[b]] ) ) )
//
// Streaming pass over Z (419 MB -> ~18us floor @ 23.3 TB/s HBM), pipelined
// 3-deep into a 4-buffer LDS ring with gfx1250 async-to-LDS loads
// (global_load_async_to_lds_b128 / ASYNCcnt / s_wait_asynccnt <= 2), so the
// branch-free top-6 min/max VALU chain overlaps the HBM stream per-wave, not
// just via occupancy.
// ---------------------------------------------------------------------------

#define TOPK            6        // K+1 with K=5
#define EPSILON_F       0.1f
#define SCALE_F         30.0f
#define THREADS         256
#define CPT             4        // columns (n values) per thread
#define COLS_PER_BLOCK  (THREADS * CPT)   // 1024
#define NBUF            4        // LDS ring depth (3 rows in flight)

typedef int v4i __attribute__((vector_size(16)));
typedef __attribute__((address_space(1))) v4i as1_v4i;
typedef __attribute__((address_space(3))) v4i as3_v4i;
typedef __attribute__((address_space(3))) void as3_void;

__device__ static inline void async_copy_b128(const float* gsrc, float* ldst) {
#if __has_builtin(__builtin_amdgcn_global_load_async_to_lds_b128)
  __builtin_amdgcn_global_load_async_to_lds_b128(
      (as1_v4i*)gsrc, (as3_v4i*)ldst, /*offset=*/0, /*cpol=*/0);
#else
  unsigned lds_off = (unsigned)(__UINTPTR_TYPE__)(as3_void*)ldst;
  asm volatile("global_load_async_to_lds_b128 %0, %1, off"
               :: "v"(lds_off), "v"(gsrc) : "memory");
#endif
}

template <int CNT>
__device__ static inline void wait_async() {
#if __has_builtin(__builtin_amdgcn_s_wait_asynccnt)
  __builtin_amdgcn_s_wait_asynccnt(CNT);
#else
  asm volatile("s_wait_asynccnt %0" :: "i"(CNT) : "memory");
#endif
}

// ---------------------------------------------------------------------------
// Kernel 1: per-(b, n-tile) sum over n of the 6th-largest-over-d value.
// grid = (B, nTilesN), block = 256 threads, each thread owns 4 n-columns.
// ---------------------------------------------------------------------------
__global__ __launch_bounds__(THREADS)
void topk6_partial_kernel(const float* __restrict__ s,
                          const float* __restrict__ Z,
                          float* __restrict__ partials,
                          int B, int D, int N) {
  __shared__ __align__(16) float zbuf[NBUF][COLS_PER_BLOCK];  // 16 KB ring
  __shared__ float srow[128];
  __shared__ float red[THREADS];

  const int b     = blockIdx.x;
  const int tileN = blockIdx.y;
  const int n0    = tileN * COLS_PER_BLOCK;
  const int tid   = threadIdx.x;

  // Stage the s[b, :] row (100 floats) into LDS once.
  for (int d = tid; d < D; d += THREADS)
    srow[d] = s[(size_t)b * D + d];

  const float* zb = Z + ((size_t)b * D) * (size_t)N + n0;

  // Prologue: put 3 rows of Z[b] in flight (4 KB contiguous each, 16B/lane).
#pragma unroll
  for (int r = 0; r < NBUF - 1; ++r)
    if (r < D)
      async_copy_b128(zb + (size_t)r * N + (size_t)tid * CPT,
                      &zbuf[r][tid * CPT]);

  // Four independent sorted top-6 register files (descending order).
  float t[CPT][TOPK];
#pragma unroll
  for (int j = 0; j < CPT; ++j)
#pragma unroll
    for (int i = 0; i < TOPK; ++i) t[j][i] = -INFINITY;

  for (int d = 0; d < D; ++d) {
    // Oldest outstanding row (d) landed for THIS wave; rows d+1, d+2 may
    // still be in flight. Barrier publishes all waves' row-d data.
    if (d < D - (NBUF - 1)) wait_async<NBUF - 2>();
    else                    wait_async<0>();
    __syncthreads();

    // Refill the slot read in iteration d-1; every wave passed the barrier
    // above only after finishing those reads, so the overwrite is safe.
    if (d + NBUF - 1 < D)
      async_copy_b128(zb + (size_t)(d + NBUF - 1) * N + (size_t)tid * CPT,
                      &zbuf[(d + NBUF - 1) & (NBUF - 1)][tid * CPT]);

    const float sd = srow[d];                       // LDS broadcast read
    const float4 z =
        *reinterpret_cast<const float4*>(&zbuf[d & (NBUF - 1)][tid * CPT]);
    const float v4[CPT] = {z.x, z.y, z.z, z.w};

#pragma unroll
    for (int j = 0; j < CPT; ++j) {
      float a = fmaf(EPSILON_F, v4[j], sd);          // s[b,d] + 0.1*Z
      // Branch-free bubble insert: keeps t[j][0] >= ... >= t[j][5].
#pragma unroll
      for (int i = 0; i < TOPK; ++i) {
        const float hi = fmaxf(t[j][i], a);
        a        = fminf(t[j][i], a);
        t[j][i]  = hi;
      }
    }
  }

  // t[j][5] is the (K+1)-th largest; sum this thread's 4 columns.
  float local = t[0][TOPK - 1] + t[1][TOPK - 1] + t[2][TOPK - 1] + t[3][TOPK - 1];
  red[tid] = local;
  __syncthreads();
#pragma unroll
  for (int off = THREADS / 2; off > 0; off >>= 1) {
    if (tid < off) red[tid] += red[tid + off];
    __syncthreads();
  }
  if (tid == 0)
    partials[(size_t)b * gridDim.y + tileN] = red[0];  // sum over this n-tile
}

// ---------------------------------------------------------------------------
// Kernel 2: epilogue.  skp1[b] = sum_partials/N, then margin/relu/mean.
// ---------------------------------------------------------------------------
__global__ __launch_bounds__(256)
void finalize_kernel(const float* __restrict__ s,
                     const int*   __restrict__ y,
                     const float* __restrict__ m_list,
                     const float* __restrict__ partials,
                     float* __restrict__ out,
                     int B, int D, int N, int nTiles) {
  __shared__ float red[256];
  float acc = 0.0f;
  for (int b = threadIdx.x; b < B; b += 256) {
    float ssum = 0.0f;
    for (int t = 0; t < nTiles; ++t) ssum += partials[(size_t)b * nTiles + t];
    const float skp1    = ssum / (float)N;
    const int   yb      = y[b];
    const float margin  = m_list[yb];
    const float correct = s[(size_t)b * D + yb];
    const float num     = SCALE_F * (margin + skp1 - correct);
    acc += fmaxf(num, 0.0f);
  }
  red[threadIdx.x] = acc;
  __syncthreads();
#pragma unroll
  for (int off = 128; off > 0; off >>= 1) {
    if (threadIdx.x < off) red[threadIdx.x] += red[threadIdx.x + off];
    __syncthreads();
  }
  if (threadIdx.x == 0) out[0] = red[0] / (float)B;
}

// ---------------------------------------------------------------------------
extern "C" void kernel_launch(void* const* d_in, const int* in_sizes, int n_in,
                              void* d_out, int out_size, void* d_ws, size_t ws_size,
                              hipStream_t stream) {
  const float* s      = (const float*)d_in[0];   // [B, D] f32
  const int*   y      = (const int*)  d_in[1];   // [B]    i32
  const float* m_list = (const float*)d_in[2];   // [D]    f32
  const float* Z      = (const float*)d_in[3];   // [B, D, N] f32

  const int B = in_sizes[1];
  const int D = in_sizes[2];
  const int N = (int)((long long)in_sizes[3] / ((long long)B * (long long)D));
  const int nTiles = (N + COLS_PER_BLOCK - 1) / COLS_PER_BLOCK;

  float* partials = (float*)d_ws;                // B * nTiles floats

  dim3 grid((unsigned)B, (unsigned)nTiles);
  topk6_partial_kernel<<<grid, THREADS, 0, stream>>>(s, Z, partials, B, D, N);
  finalize_kernel<<<1, 256, 0, stream>>>(s, y, m_list, partials,
                                         (float*)d_out, B, D, N, nTiles);
}